// RejectionSamplerPatch_37967510896989
// MI455X (gfx1250) — compile-verified
//
#include <hip/hip_runtime.h>
#include <stdint.h>

// Problem constants (match reference): B=128, K=8, V=32000.
#define B_   128
#define K_   8
#define V_   32000
#define TPB  256
#define VEC4_PER_ROW (V_ / 4)                       // 8000 float4s per row
#define NTILES ((VEC4_PER_ROW + TPB - 1) / TPB)     // 32 tiles of 256 vec4

typedef __attribute__((ext_vector_type(4))) float v4f;

// ---------- deterministic stateless PRNG (per call, per element) ----------
__device__ __forceinline__ uint32_t hash32(uint32_t x) {
  x ^= x >> 16; x *= 0x7feb352dU;
  x ^= x >> 15; x *= 0x846ca68bU;
  x ^= x >> 16;
  return x;
}
__device__ __forceinline__ float u01(uint32_t h) {
  // (0,1): 24-bit mantissa uniform, offset by half an ulp so log() is finite
  return (float)(h >> 8) * (1.0f / 16777216.0f) + (0.5f / 16777216.0f);
}

// ---------- CDNA5 async global -> LDS (ASYNCcnt path) ----------
__device__ __forceinline__ void async_ld_b128(uint32_t lds_off, const void* g) {
  // vdst = LDS byte address (VGPR), vaddr = 64-bit global address, GV mode
  asm volatile("global_load_async_to_lds_b128 %0, %1, off"
               :: "v"(lds_off), "v"(g)
               : "memory");
}
template <int N>
__device__ __forceinline__ void wait_async() {
  asm volatile("s_wait_asynccnt %0" :: "i"(N) : "memory");
}

// =====================================================================
// Kernel 1: one workgroup per (b,k) row.
//  - accept test (gather) on lane 0
//  - recovered token = argmax_v f_v / E_v   (exponential race == categorical)
//    computed in a single streamed pass, double-buffered through LDS.
// =====================================================================
__global__ void __launch_bounds__(TPB)
rs_score_kernel(const float* __restrict__ twb,   // (B, K+1, V) target w/ bonus
                const float* __restrict__ dp,    // (B, K, V)   draft probs
                const int*   __restrict__ dtok,  // (B, K)      draft token ids
                int* __restrict__ ws_acc,        // (B*K) accepted flags
                int* __restrict__ ws_rec)        // (B*K) recovered token ids
{
  __shared__ v4f tT[2][TPB];   // 8 KB  target tiles (double buffered)
  __shared__ v4f tD[2][TPB];   // 8 KB  draft tiles
  __shared__ float sF[TPB / 32], sE[TPB / 32];
  __shared__ int   sV[TPB / 32];

  const int r   = blockIdx.x;          // row id in [0, B*K)
  const int b   = r / K_;
  const int k   = r - b * K_;
  const int tid = threadIdx.x;

  const float* trow = twb + ((size_t)b * (K_ + 1) + (size_t)k) * (size_t)V_;
  const float* drow = dp  + (size_t)r * (size_t)V_;

  // ---- acceptance test: u < min(t[tok]/d[tok], 1) ----
  if (tid == 0) {
    const int tok = dtok[r];
    const float st = trow[tok];
    const float sd = drow[tok];
    const float ratio = fminf(st / sd, 1.0f);
    const float ua = u01(hash32(0xACCE5517u ^ (uint32_t)(r * 0x9E3779B9u)));
    ws_acc[r] = (ua < ratio) ? 1 : 0;
  }

  // LDS byte offsets of this thread's private 16B slots
  const uint32_t myT0 = (uint32_t)(size_t)&tT[0][tid];
  const uint32_t myT1 = (uint32_t)(size_t)&tT[1][tid];
  const uint32_t myD0 = (uint32_t)(size_t)&tD[0][tid];
  const uint32_t myD1 = (uint32_t)(size_t)&tD[1][tid];

  // Prologue: issue tile 0 (clamped index => every wave always issues 2 ops)
  {
    const int i0 = min(tid, VEC4_PER_ROW - 1);
    async_ld_b128(myT0, trow + (size_t)i0 * 4);
    async_ld_b128(myD0, drow + (size_t)i0 * 4);
  }

  const float TINY = 1.17549435e-38f;  // finfo(float32).tiny
  float bf = 0.0f, bE = 1.0f;          // race state: best f, best E
  int   bv = 0;
  const uint32_t hbase = (uint32_t)(r * 0x01000193u) ^ 0xB5297A4Du;

  for (int i = 0; i < NTILES; ++i) {
    const int buf = i & 1;
    if (i + 1 < NTILES) {
      const int ni = min((i + 1) * TPB + tid, VEC4_PER_ROW - 1);
      async_ld_b128(buf ? myT0 : myT1, trow + (size_t)ni * 4);
      async_ld_b128(buf ? myD0 : myD1, drow + (size_t)ni * 4);
      wait_async<2>();   // current tile's 2 loads complete (in-order)
    } else {
      wait_async<0>();
    }

    const v4f tv = tT[buf][tid];       // ds_load_b128, own slot only
    const v4f dv = tD[buf][tid];
    const int v4 = min(i * TPB + tid, VEC4_PER_ROW - 1);
#pragma unroll
    for (int j = 0; j < 4; ++j) {
      const int   v = v4 * 4 + j;
      const float f = fmaxf(tv[j] - dv[j], TINY);
      const float u = u01(hash32(hbase ^ (uint32_t)(v * 0x85EBCA6Bu)));
      const float E = -__logf(u);      // Exp(1)
      // f/E > bf/bE  <=>  f*bE > bf*E  (all positive); strict => clamp dups safe
      if (f * bE > bf * E) { bf = f; bE = E; bv = v; }
    }
  }

  // ---- wave32 butterfly reduction of the race max ----
#pragma unroll
  for (int off = 16; off > 0; off >>= 1) {
    const float of = __shfl_xor(bf, off, 32);
    const float oE = __shfl_xor(bE, off, 32);
    const int   ov = __shfl_xor(bv, off, 32);
    const float a = of * bE, c = bf * oE;
    if (a > c || (a == c && ov < bv)) { bf = of; bE = oE; bv = ov; }
  }

  const int wid = tid >> 5;
  if ((tid & 31) == 0) { sF[wid] = bf; sE[wid] = bE; sV[wid] = bv; }
  __syncthreads();
  if (tid == 0) {
    float Bf = sF[0], BE = sE[0];
    int Bv = sV[0];
    for (int w = 1; w < TPB / 32; ++w) {
      const float a = sF[w] * BE, c = Bf * sE[w];
      if (a > c || (a == c && sV[w] < Bv)) { Bf = sF[w]; BE = sE[w]; Bv = sV[w]; }
    }
    ws_rec[r] = Bv;
  }
}

// =====================================================================
// Kernel 2: per-batch limits + output assembly, (B, K+1) int32
// =====================================================================
__global__ void rs_finalize_kernel(const int* __restrict__ ws_acc,
                                   const int* __restrict__ ws_rec,
                                   const int* __restrict__ dtok,
                                   const int* __restrict__ bonus,
                                   int* __restrict__ out)
{
  const int b = blockIdx.x * blockDim.x + threadIdx.x;
  if (b >= B_) return;

  int limit = K_;
  for (int k = 0; k < K_; ++k) {
    if (!ws_acc[b * K_ + k]) { limit = k; break; }
  }
#pragma unroll
  for (int k = 0; k < K_; ++k) {
    int v;
    if (k < limit)       v = dtok[b * K_ + k];
    else if (k == limit) v = ws_rec[b * K_ + k];
    else                 v = -1;
    out[b * (K_ + 1) + k] = v;
  }
  out[b * (K_ + 1) + K_] = (limit == K_) ? bonus[b] : -1;
}

extern "C" void kernel_launch(void* const* d_in, const int* in_sizes, int n_in,
                              void* d_out, int out_size, void* d_ws, size_t ws_size,
                              hipStream_t stream) {
  (void)in_sizes; (void)n_in; (void)out_size; (void)ws_size;
  const float* twb  = (const float*)d_in[0];  // target_with_bonus_probs (B,K+1,V)
  const int*   bon  = (const int*)  d_in[1];  // bonus_token_ids        (B,1)
  const float* dp   = (const float*)d_in[2];  // draft_probs            (B,K,V)
  const int*   dtk  = (const int*)  d_in[3];  // draft_token_ids        (B,K)
  int* out = (int*)d_out;                      // (B, K+1) int32

  int* ws_acc = (int*)d_ws;                    // B*K flags
  int* ws_rec = ws_acc + B_ * K_;              // B*K recovered ids

  rs_score_kernel<<<B_ * K_, TPB, 0, stream>>>(twb, dp, dtk, ws_acc, ws_rec);
  rs_finalize_kernel<<<1, B_, 0, stream>>>(ws_acc, ws_rec, dtk, bon, out);
}